// PoolWeightedSum_71665824301162
// MI455X (gfx1250) — compile-verified
//
#include <hip/hip_runtime.h>

// PoolWeightedSum on MI455X (gfx1250, wave32).
// HBM-bandwidth-bound (102.4 MB feats read, ~4.5 us floor at 23.3 TB/s).
// feats is streamed HBM -> LDS exactly once via GLOBAL_LOAD_ASYNC_TO_LDS_B128
// (ASYNCcnt-tracked, double-buffered to overlap with compute), and the per-graph
// 4x256 segment-sum-of-outer-products is a K-reduction GEMM on
// V_WMMA_F32_16X16X4_F32 (full f32 precision, matches reference math).

typedef float v2f __attribute__((ext_vector_type(2)));
typedef float v8f __attribute__((ext_vector_type(8)));

#define NGRAPHS 512
#define FDIM    256
#define ODIM    4
#define FSTR    260      // padded LDS row stride in floats: bank(4i + k) -> conflict-free
#define CHUNK   32       // nodes staged per iteration
#define BLOCK   512      // 16 wave32s; wave w owns feature columns [16w, 16w+16)

__global__ __launch_bounds__(BLOCK)
void pool_weighted_sum_kernel(const float* __restrict__ feats,
                              const int*   __restrict__ batch,
                              const float* __restrict__ W,
                              const float* __restrict__ bias,
                              float*       __restrict__ out,
                              int N)
{
    __shared__ __align__(16) float lds_feats[2][CHUNK * FSTR];  // double-buffered 32x256
    __shared__ __align__(16) float lds_W[ODIM * FSTR];          // 4x256 (padded)
    __shared__ float lds_gate[CHUNK * ODIM];                    // 32x4, node-major
    __shared__ int   seg[2];

    const int t     = threadIdx.x;
    const int g     = blockIdx.x;          // graph id
    const int wave  = t >> 5;
    const int lane  = t & 31;
    const int m     = lane & 15;           // M row (A) or N column (B/D) within tile
    const int khalf = lane >> 4;           // 0 = lanes 0-15, 1 = lanes 16-31
    const unsigned long long fbase = (unsigned long long)(uintptr_t)feats;

    // ---- segment bounds: lower_bound(batch, g) / lower_bound(batch, g+1) ----
    if (t == 0) {
        int lo = 0, hi = N;
        while (lo < hi) { int mid = (lo + hi) >> 1; if (batch[mid] < g)     lo = mid + 1; else hi = mid; }
        seg[0] = lo;
        hi = N;
        while (lo < hi) { int mid = (lo + hi) >> 1; if (batch[mid] < g + 1) lo = mid + 1; else hi = mid; }
        seg[1] = lo;
    }
    // ---- preload W (4x256) into padded LDS ----
    for (int q = t; q < ODIM * FDIM; q += BLOCK) {
        int o = q >> 8, k = q & 255;
        lds_W[o * FSTR + k] = W[q];
    }
    __syncthreads();
    const int s = seg[0], e = seg[1];

    // Issue one chunk's HBM->LDS transfer: async B128 per valid float4 slot,
    // ds_store zeros for tail rows (disjoint addresses -> no ordering hazard).
    auto issue_chunk = [&](int cbase, int buf) {
        #pragma unroll
        for (int j = 0; j < (CHUNK * FDIM / 4) / BLOCK; ++j) {   // 4 slots/thread
            int q   = t + j * BLOCK;          // 0..2047 float4 slots
            int row = q >> 6;                 // 64 float4 per 256-float row
            int c4  = q & 63;
            int gi  = cbase + row;
            float* ldst = &lds_feats[buf][row * FSTR + (c4 << 2)];
            if (gi < e) {
                unsigned loff = (unsigned)(uintptr_t)ldst;       // LDS byte address
                unsigned goff = (unsigned)(gi * (FDIM * 4) + (c4 << 4)); // byte offset
                asm volatile("global_load_async_to_lds_b128 %0, %1, %2"
                             :: "v"(loff), "v"(goff), "s"(fbase) : "memory");
            } else {
                *(float4*)ldst = make_float4(0.f, 0.f, 0.f, 0.f);
            }
        }
    };

    v8f acc = {0.f, 0.f, 0.f, 0.f, 0.f, 0.f, 0.f, 0.f};
    const int colbase = wave << 4;

    if (s < e) issue_chunk(s, 0);           // prologue: prefetch chunk 0
    int buf = 0;
    for (int base = s; base < e; base += CHUNK) {
        asm volatile("s_wait_asynccnt 0x0" ::: "memory");  // this wave's chunk landed
        __syncthreads();                                   // all waves' chunks landed

        if (base + CHUNK < e) issue_chunk(base + CHUNK, buf ^ 1);  // overlap next load

        // -- gates for 32 nodes x 4 outputs (128 threads, full dots from LDS) --
        if (t < CHUNK * ODIM) {
            int i = t >> 2, o = t & 3;
            const float* fr = &lds_feats[buf][i * FSTR];
            const float* wr = &lds_W[o * FSTR];
            float sum = 0.f;
            #pragma unroll 8
            for (int k = 0; k < FDIM; ++k) sum = __builtin_fmaf(fr[k], wr[k], sum);
            float gate = 0.f;
            if (base + i < e) gate = 1.f / (1.f + __expf(-(sum + bias[o])));
            lds_gate[i * ODIM + o] = gate;
        }
        __syncthreads();

        // -- D(16x16) += A(16x4) x B(4x16), 8 K-steps over the 32 staged nodes --
        //    A[m][k] = gate[node k][out m] (m<4 else 0); B[k][n] = feats[node k][col n].
        //    f32 A layout: VGPR0 = K {k0 | k0+2}, VGPR1 = K {k0+1 | k0+3} (low|high lanes).
        #pragma unroll
        for (int k0 = 0; k0 < CHUNK; k0 += 4) {
            int ka = k0 + (khalf << 1);
            v2f a, bf;
            a.x  = (m < ODIM) ? lds_gate[(ka + 0) * ODIM + m] : 0.f;
            a.y  = (m < ODIM) ? lds_gate[(ka + 1) * ODIM + m] : 0.f;
            bf.x = lds_feats[buf][(ka + 0) * FSTR + colbase + m];
            bf.y = lds_feats[buf][(ka + 1) * FSTR + colbase + m];
            acc = __builtin_amdgcn_wmma_f32_16x16x4_f32(
                      /*neg_a=*/false, a, /*neg_b=*/false, bf,
                      /*c_mod=*/(short)0, acc, /*reuse_a=*/false, /*reuse_b=*/false);
        }
        buf ^= 1;
        // no bottom barrier needed: next iteration's wait+barrier precedes any LDS write
    }

    // -- Store: D VGPR o, lanes 0-15 hold M=o, N=lane -> out[g, o*256 + colbase + lane].
    if (khalf == 0) {
        float* dst = out + g * (ODIM * FDIM) + colbase + m;
        #pragma unroll
        for (int o = 0; o < ODIM; ++o) dst[o * FDIM] = acc[o];
    }
}

extern "C" void kernel_launch(void* const* d_in, const int* in_sizes, int n_in,
                              void* d_out, int out_size, void* d_ws, size_t ws_size,
                              hipStream_t stream) {
    const float* feats = (const float*)d_in[0];   // [N, 256]
    const int*   batch = (const int*)d_in[1];     // [N] sorted graph ids
    const float* W     = (const float*)d_in[2];   // [4, 256]
    const float* bias  = (const float*)d_in[3];   // [4]
    float*       out   = (float*)d_out;           // [512, 1024]
    const int N = in_sizes[1];

    pool_weighted_sum_kernel<<<NGRAPHS, BLOCK, 0, stream>>>(feats, batch, W, bias, out, N);
}